// ReservoirNet_27401891349223
// MI455X (gfx1250) — compile-verified
//
#include <hip/hip_runtime.h>

// Leaky-integrator ESN on gfx1250:
//   x_t = (1-A)*x_{t-1} + A*tanh(Win@[1;u_t] + W@x_{t-1})
// Single persistent kernel, per-step grid barrier, fp32 WMMA (16x16x4).
// Each wave owns TWO 16-column N-tiles so every A (x) load feeds 2 WMMAs,
// halving x traffic from L2; W is still touched exactly once per step.

#define T_STEPS 1024
#define BATCH   32
#define INSZ    128
#define RES     2048
#define LEAK    0.3f
#define NT      2                       // N-tiles per block/wave
#define NBLOCKS (RES / (16 * NT))       // 64 blocks, one per 32-column slab
#define NWAVES  8
#define THREADS (NWAVES * 32)

typedef float v2f __attribute__((ext_vector_type(2)));
typedef float v8f __attribute__((ext_vector_type(8)));

__device__ __forceinline__ v8f wmma_f32(v2f a, v2f b, v8f c) {
    // D = A(16x4) * B(4x16) + C(16x16), full fp32
    return __builtin_amdgcn_wmma_f32_16x16x4_f32(false, a, false, b,
                                                 (short)0, c, false, false);
}

__global__ void esn_init_counters(int* counters) {
    int i = blockIdx.x * blockDim.x + threadIdx.x;
    if (i < T_STEPS) counters[i] = 0;
}

__launch_bounds__(THREADS)
__global__ void esn_persistent(const float* __restrict__ u,
                               const float* __restrict__ Win,
                               const float* __restrict__ W,
                               float* __restrict__ states,
                               int* __restrict__ counters) {
    // per-wave partial tiles: [wave][tile(NT)][m(32) x n(16)]
    __shared__ float lds[NWAVES * NT * BATCH * 16];   // 32 KB

    const int tid   = threadIdx.x;
    const int lane  = tid & 31;
    const int wave  = tid >> 5;
    const int nbase = blockIdx.x * (16 * NT);    // first column of this block's slab
    const int row   = lane & 15;                 // M-row (A) or N-col (B) index
    const int koff  = (lane >> 4) * 2;           // lanes>=16 carry K+2,K+3

    // B operands: output columns nbase+row and nbase+16+row => rows of W / Win
    const float* Wrow0   = W   + (size_t)(nbase + row) * RES;
    const float* Wrow1   = W   + (size_t)(nbase + 16 + row) * RES;
    const float* WinRow0 = Win + (size_t)(nbase + row) * (INSZ + 1);
    const float* WinRow1 = Win + (size_t)(nbase + 16 + row) * (INSZ + 1);

    for (int t = 0; t < T_STEPS; ++t) {
        v8f acc00 = {0.f,0.f,0.f,0.f,0.f,0.f,0.f,0.f};  // tile0, batch 0..15
        v8f acc01 = {0.f,0.f,0.f,0.f,0.f,0.f,0.f,0.f};  // tile0, batch 16..31
        v8f acc10 = {0.f,0.f,0.f,0.f,0.f,0.f,0.f,0.f};  // tile1, batch 0..15
        v8f acc11 = {0.f,0.f,0.f,0.f,0.f,0.f,0.f,0.f};  // tile1, batch 16..31

        // ---- input projection: K = 128 split across 8 waves (16 K each) ----
        {
            const float* ut = u + (size_t)t * BATCH * INSZ;
            const int kb = wave * (INSZ / NWAVES);
            #pragma unroll
            for (int i = 0; i < (INSZ / NWAVES) / 4; ++i) {
                int k = kb + 4 * i + koff;
                v2f a0, a1, b0, b1;
                a0.x = ut[row * INSZ + k];
                a0.y = ut[row * INSZ + k + 1];
                a1.x = ut[(16 + row) * INSZ + k];
                a1.y = ut[(16 + row) * INSZ + k + 1];
                b0.x = WinRow0[1 + k];           // row stride 129 (odd) -> scalar loads
                b0.y = WinRow0[1 + k + 1];
                b1.x = WinRow1[1 + k];
                b1.y = WinRow1[1 + k + 1];
                acc00 = wmma_f32(a0, b0, acc00);
                acc01 = wmma_f32(a1, b0, acc01);
                acc10 = wmma_f32(a0, b1, acc10);
                acc11 = wmma_f32(a1, b1, acc11);
            }
        }

        // ---- recurrent part: K = 2048 split across 8 waves (256 K each) ----
        if (t > 0) {
            const float* xprev = states + (size_t)(t - 1) * BATCH * RES;
            const int kb = wave * (RES / NWAVES);
            #pragma unroll 4
            for (int i = 0; i < (RES / NWAVES) / 4; ++i) {
                int k = kb + 4 * i + koff;
                v2f a0 = *(const v2f*)(xprev + (size_t)row * RES + k);
                v2f a1 = *(const v2f*)(xprev + (size_t)(16 + row) * RES + k);
                v2f b0 = *(const v2f*)(Wrow0 + k);   // each W element: once per step
                v2f b1 = *(const v2f*)(Wrow1 + k);
                acc00 = wmma_f32(a0, b0, acc00);
                acc01 = wmma_f32(a1, b0, acc01);
                acc10 = wmma_f32(a0, b1, acc10);
                acc11 = wmma_f32(a1, b1, acc11);
            }
        }

        // ---- stash per-wave partial tiles to LDS (tile-major) ----
        {
            float* p = lds + wave * (NT * BATCH * 16);
            const int mhalf = (lane < 16) ? 0 : 8;   // C/D layout: M split per half-wave
            #pragma unroll
            for (int i = 0; i < 8; ++i) {
                p[(mhalf + i) * 16 + row]            = acc00[i];
                p[(16 + mhalf + i) * 16 + row]       = acc01[i];
                p[512 + (mhalf + i) * 16 + row]      = acc10[i];
                p[512 + (16 + mhalf + i) * 16 + row] = acc11[i];
            }
        }
        __syncthreads();

        // ---- deterministic reduction + bias + tanh + leaky update + store ----
        {
            float*       xout = states + (size_t)t * BATCH * RES;
            const float* xold = states + (size_t)(t - 1) * BATCH * RES;
            #pragma unroll
            for (int e2 = 0; e2 < NT * BATCH * 16 / THREADS; ++e2) {
                int e    = e2 * THREADS + tid;       // 1024 outputs, coalesced
                int tile = e >> 9;                   // 0 or 1
                int ei   = e & 511;
                int m    = ei >> 4;                  // batch row 0..31
                int n    = ei & 15;                  // column within tile
                int col  = nbase + tile * 16 + n;    // global column
                float s = 0.f;
                #pragma unroll
                for (int w = 0; w < NWAVES; ++w)     // fixed order => deterministic
                    s += lds[w * (NT * BATCH * 16) + e];
                s += Win[(size_t)col * (INSZ + 1)];  // bias column of Win
                float xo = (t > 0) ? xold[(size_t)m * RES + col] : 0.f;
                xout[(size_t)m * RES + col] = (1.0f - LEAK) * xo + LEAK * tanhf(s);
            }
        }
        __syncthreads();   // protect LDS before next step overwrites it

        // ---- grid-wide barrier: all 64 blocks must finish step t ----
        if (tid == 0) {
            __threadfence();                 // make states[t] visible device-wide
            atomicAdd(&counters[t], 1);
            while (__hip_atomic_load(&counters[t], __ATOMIC_RELAXED,
                                     __HIP_MEMORY_SCOPE_AGENT) < NBLOCKS) {
                __builtin_amdgcn_s_sleep(1);
            }
            __threadfence();                 // acquire: see other blocks' states[t]
        }
        __syncthreads();
    }
}

extern "C" void kernel_launch(void* const* d_in, const int* in_sizes, int n_in,
                              void* d_out, int out_size, void* d_ws, size_t ws_size,
                              hipStream_t stream) {
    const float* u   = (const float*)d_in[0];   // [1024, 32, 128]
    const float* Win = (const float*)d_in[1];   // [2048, 129]
    const float* W   = (const float*)d_in[2];   // [2048, 2048]
    float* states    = (float*)d_out;           // [1024, 32, 2048]
    int*   counters  = (int*)d_ws;              // 1024 ints of scratch

    esn_init_counters<<<(T_STEPS + 255) / 256, 256, 0, stream>>>(counters);
    esn_persistent<<<NBLOCKS, THREADS, 0, stream>>>(u, Win, W, states, counters);
}